// AdvanceGNNModel_20882130993665
// MI455X (gfx1250) — compile-verified
//
#include <hip/hip_runtime.h>

typedef __attribute__((ext_vector_type(2))) float v2f;
typedef __attribute__((ext_vector_type(8))) float v8f;

#define NEG_SLOPE 0.2f

// For slope in (0,1): leaky_relu(x) == max(x, slope*x).
// Compiles to v_pk_mul_f32 + dual v_max_num_f32 (no cmp/cndmask chains).
__device__ __forceinline__ float lrelu(float v) {
    return fmaxf(v, v * NEG_SLOPE);
}

// ---------------- degree / normalization ----------------

__global__ void fill_ones_kernel(float* __restrict__ p, int n) {
    int i = blockIdx.x * blockDim.x + threadIdx.x;
    if (i < n) p[i] = 1.0f;   // self-loop contributes 1 to every degree
}

__global__ void count_deg_kernel(const int* __restrict__ dst, float* __restrict__ deg, int E) {
    int i = blockIdx.x * blockDim.x + threadIdx.x;
    if (i < E) unsafeAtomicAdd(&deg[dst[i]], 1.0f);
}

__global__ void to_rsqrt_kernel(float* __restrict__ p, int n) {
    int i = blockIdx.x * blockDim.x + threadIdx.x;
    if (i < n) {
        float d = p[i];
        p[i] = d > 0.0f ? rsqrtf(d) : 0.0f;
    }
}

// ---------------- WMMA GEMM: out[N,64] = act(A[N,64]) @ W[64,64] ----------------
// One wave computes a 16x16 tile via 16 chained v_wmma_f32_16x16x4_f32.
// Block = 128 threads = 4 waves = 4 column tiles (full 64 output cols).
// A fragment (16x4 f32): lanes 0-15 -> K=k+0,k+1 ; lanes 16-31 -> K=k+2,k+3 (ISA 7.12.2).
// B fragment (4x16 f32): mirrored K split across lane halves.
// C/D (16x16 f32): VGPR r holds row r (lanes 0-15) / row r+8 (lanes 16-31).

template <int APPLY_RELU>
__global__ void __launch_bounds__(128)
gemm64_wmma_kernel(const float* __restrict__ A, const float* __restrict__ W,
                   float* __restrict__ out) {
    const int lane = threadIdx.x & 31;
    const int wave = threadIdx.x >> 5;
    const int m0   = blockIdx.x * 16;
    const int n0   = wave * 16;
    const int mr   = lane & 15;
    const int hi   = lane >> 4;             // 0: K pair 0,1 ; 1: K pair 2,3
    const float* arow = A + (size_t)(m0 + mr) * 64 + hi * 2;

    v8f c = {};
    #pragma unroll
    for (int k = 0; k < 64; k += 4) {
        float a0 = arow[k + 0];
        float a1 = arow[k + 1];
        if (APPLY_RELU) { a0 = lrelu(a0); a1 = lrelu(a1); }
        v2f a = {a0, a1};
        const int kb = k + hi * 2;
        v2f b = {W[(size_t)(kb + 0) * 64 + n0 + mr],
                 W[(size_t)(kb + 1) * 64 + n0 + mr]};
        c = __builtin_amdgcn_wmma_f32_16x16x4_f32(
                /*neg_a=*/false, a, /*neg_b=*/false, b,
                /*c_mod=*/(short)0, c, /*reuse_a=*/false, /*reuse_b=*/false);
    }

    float* orow = out + (size_t)(m0 + hi * 8) * 64 + n0 + mr;
    #pragma unroll
    for (int r = 0; r < 8; ++r) orow[(size_t)r * 64] = c[r];
}

// ---------------- WMMA GEMM: out[N,4] = lrelu(A[N,64]) @ W[64,4] ----------------
// Only 4 valid output columns. B loads are made unconditional (column index
// clamped with mr&3, always in-bounds of the 64x4 W) and invalid lanes are
// zeroed by one hoisted mask multiply -- no divergent exec-mask branches in
// the inner loop.
// Block = 128 threads = 4 waves = 4 consecutive 16-row tiles.

__global__ void __launch_bounds__(128)
gemm4_wmma_kernel(const float* __restrict__ A, const float* __restrict__ W,
                  float* __restrict__ out, int n_rows) {
    const int lane = threadIdx.x & 31;
    const int wave = threadIdx.x >> 5;
    const int m0   = (blockIdx.x * 4 + wave) * 16;
    if (m0 >= n_rows) return;               // wave-uniform: EXEC stays all-ones
    const int mr  = lane & 15;
    const int hi  = lane >> 4;
    const int nc  = mr & 3;                 // clamped column index (in-bounds)
    const float msk = (mr < 4) ? 1.0f : 0.0f;
    const float* arow = A + (size_t)(m0 + mr) * 64 + hi * 2;

    v8f c = {};
    #pragma unroll
    for (int k = 0; k < 64; k += 4) {
        v2f a = {lrelu(arow[k + 0]), lrelu(arow[k + 1])};
        const int kb = k + hi * 2;
        v2f b = {W[(size_t)(kb + 0) * 4 + nc] * msk,
                 W[(size_t)(kb + 1) * 4 + nc] * msk};
        c = __builtin_amdgcn_wmma_f32_16x16x4_f32(
                false, a, false, b, (short)0, c, false, false);
    }

    if (mr < 4) {
        float* orow = out + (size_t)(m0 + hi * 8) * 4 + mr;
        #pragma unroll
        for (int r = 0; r < 8; ++r) orow[(size_t)r * 4] = c[r];
    }
}

// ---------------- epilogue init: acc = bias + tmp * dinv^2 (self-loop fused) ----------------

__global__ void init_acc64_kernel(const float* __restrict__ tmp, const float* __restrict__ bias,
                                  const float* __restrict__ dinv, float* __restrict__ acc, int N) {
    int t = blockIdx.x * blockDim.x + threadIdx.x;     // N*16 threads, float4 each
    if (t >= N * 16) return;
    int row = t >> 4;
    int f   = (t & 15) << 2;
    float di = dinv[row];
    float w  = di * di;
    float4 h = *(const float4*)(tmp + (size_t)row * 64 + f);
    float4 b = *(const float4*)(bias + f);
    float4 o;
    o.x = fmaf(h.x, w, b.x);
    o.y = fmaf(h.y, w, b.y);
    o.z = fmaf(h.z, w, b.z);
    o.w = fmaf(h.w, w, b.w);
    *(float4*)(acc + (size_t)row * 64 + f) = o;
}

__global__ void init_acc4_kernel(const float* __restrict__ tmp4, const float* __restrict__ bias,
                                 const float* __restrict__ dinv, float* __restrict__ acc, int N) {
    int i = blockIdx.x * blockDim.x + threadIdx.x;
    if (i >= N) return;
    float di = dinv[i];
    float w  = di * di;
    float4 h = *(const float4*)(tmp4 + (size_t)i * 4);
    float4 b = *(const float4*)(bias);
    float4 o;
    o.x = fmaf(h.x, w, b.x);
    o.y = fmaf(h.y, w, b.y);
    o.z = fmaf(h.z, w, b.z);
    o.w = fmaf(h.w, w, b.w);
    *(float4*)(acc + (size_t)i * 4) = o;
}

// ---------------- edge scatter: acc[dst] += h[src] * dinv[src]*dinv[dst] ----------------
// One thread per (edge, 4 features): B128 gather + 4 hardware f32 atomics (L2-resident).

__global__ void scatter64_kernel(const int* __restrict__ src, const int* __restrict__ dst,
                                 const float* __restrict__ dinv, const float* __restrict__ h,
                                 float* __restrict__ acc, int E) {
    int t = blockIdx.x * blockDim.x + threadIdx.x;
    int e = t >> 4;
    if (e >= E) return;
    int f = (t & 15) << 2;
    int s = src[e], d = dst[e];
    float nrm = dinv[s] * dinv[d];
    float4 hv = *(const float4*)(h + (size_t)s * 64 + f);
    float* p = acc + (size_t)d * 64 + f;
    unsafeAtomicAdd(p + 0, hv.x * nrm);
    unsafeAtomicAdd(p + 1, hv.y * nrm);
    unsafeAtomicAdd(p + 2, hv.z * nrm);
    unsafeAtomicAdd(p + 3, hv.w * nrm);
}

__global__ void scatter4_kernel(const int* __restrict__ src, const int* __restrict__ dst,
                                const float* __restrict__ dinv, const float* __restrict__ h4,
                                float* __restrict__ acc, int E) {
    int e = blockIdx.x * blockDim.x + threadIdx.x;
    if (e >= E) return;
    int s = src[e], d = dst[e];
    float nrm = dinv[s] * dinv[d];
    float4 hv = *(const float4*)(h4 + (size_t)s * 4);
    float* p = acc + (size_t)d * 4;
    unsafeAtomicAdd(p + 0, hv.x * nrm);
    unsafeAtomicAdd(p + 1, hv.y * nrm);
    unsafeAtomicAdd(p + 2, hv.z * nrm);
    unsafeAtomicAdd(p + 3, hv.w * nrm);
}

// ---------------- launcher ----------------

extern "C" void kernel_launch(void* const* d_in, const int* in_sizes, int n_in,
                              void* d_out, int out_size, void* d_ws, size_t ws_size,
                              hipStream_t stream) {
    const float* x  = (const float*)d_in[0];
    const int*   ei = (const int*)d_in[1];
    const float* W0 = (const float*)d_in[2];
    const float* b0 = (const float*)d_in[3];
    const float* W1 = (const float*)d_in[4];
    const float* b1 = (const float*)d_in[5];
    const float* W2 = (const float*)d_in[6];
    const float* b2 = (const float*)d_in[7];

    const int N = in_sizes[0] / 64;   // 50000 (multiple of 16)
    const int E = in_sizes[1] / 2;    // 800000
    const int* src = ei;
    const int* dst = ei + E;

    float* outp = (float*)d_out;

    // workspace layout (16B-aligned slices)
    float* ws   = (float*)d_ws;
    float* dinv = ws;                                  // N floats (holds deg, then rsqrt)
    float* tmp  = ws + 256 * (((size_t)N + 255) / 256);// N*64 floats (h @ W)
    float* acc  = tmp + (size_t)N * 64;                // N*64 floats (aggregated output)

    const int T = 256;
    const int rowTiles = N / 16;

    // --- symmetric normalization D^-1/2 ---
    fill_ones_kernel<<<(N + T - 1) / T, T, 0, stream>>>(dinv, N);
    count_deg_kernel<<<(E + T - 1) / T, T, 0, stream>>>(dst, dinv, E);
    to_rsqrt_kernel<<<(N + T - 1) / T, T, 0, stream>>>(dinv, N);

    // --- layer 0: x @ W0, aggregate, (+b0); leaky relu deferred into layer-1 GEMM ---
    gemm64_wmma_kernel<0><<<rowTiles, 128, 0, stream>>>(x, W0, tmp);
    init_acc64_kernel<<<(N * 16 + T - 1) / T, T, 0, stream>>>(tmp, b0, dinv, acc, N);
    scatter64_kernel<<<((size_t)E * 16 + T - 1) / T, T, 0, stream>>>(src, dst, dinv, tmp, acc, E);

    // --- layer 1: lrelu(acc) @ W1, aggregate, (+b1) ---
    gemm64_wmma_kernel<1><<<rowTiles, 128, 0, stream>>>(acc, W1, tmp);
    init_acc64_kernel<<<(N * 16 + T - 1) / T, T, 0, stream>>>(tmp, b1, dinv, acc, N);
    scatter64_kernel<<<((size_t)E * 16 + T - 1) / T, T, 0, stream>>>(src, dst, dinv, tmp, acc, E);

    // --- layer 2: lrelu(acc) @ W2 (64->4), aggregate into d_out ---
    gemm4_wmma_kernel<<<(rowTiles + 3) / 4, 128, 0, stream>>>(acc, W2, tmp, N);
    init_acc4_kernel<<<(N + T - 1) / T, T, 0, stream>>>(tmp, b2, dinv, outp, N);
    scatter4_kernel<<<(E + T - 1) / T, T, 0, stream>>>(src, dst, dinv, tmp, outp, E);
}